// QuantumConvLayer_22668837388961
// MI455X (gfx1250) — compile-verified
//
#include <hip/hip_runtime.h>
#include <hip/hip_bf16.h>
#include <stdint.h>

#define TPB 256

// ---------------------------------------------------------------------------
// 2-qubit real-amplitude circuit evaluation.
// State s[i][j]: i = qubit-1 axis (tensor axis 1), j = qubit-0 axis (axis 2).
// Circuit: ry_q0(a0) ry_q1(a1) CX ry_q0(a2) ry_q1(a3) ry_q0(th0) ry_q1(th1) CX
// Output: (s00^2 + s10^2) - (s01^2 + s11^2)   (final CX swaps within the j=1
// column, so it does not change the output).
// ---------------------------------------------------------------------------
__device__ __forceinline__ float qsim(float a0, float a1, float a2, float a3,
                                      float ct0, float st0, float ct1, float st1) {
    float c0 = __cosf(0.5f * a0), s0 = __sinf(0.5f * a0);
    float c1 = __cosf(0.5f * a1), s1 = __sinf(0.5f * a1);
    float c2 = __cosf(0.5f * a2), s2 = __sinf(0.5f * a2);
    float c3 = __cosf(0.5f * a3), s3 = __sinf(0.5f * a3);

    // |00> -> ry_q0(a0): s00=c0, s01=s0, s10=s11=0 (let const-fold do the rest)
    float x00 = c0, x01 = s0, x10 = 0.0f, x11 = 0.0f;
    float t0, t1;

    // ry_q1(a1): columns j: (x0j, x1j) -> (c*x0j - s*x1j, s*x0j + c*x1j)
    t0 = c1 * x00 - s1 * x10; t1 = s1 * x00 + c1 * x10; x00 = t0; x10 = t1;
    t0 = c1 * x01 - s1 * x11; t1 = s1 * x01 + c1 * x11; x01 = t0; x11 = t1;

    // CX: swap x01 <-> x11
    t0 = x01; x01 = x11; x11 = t0;

    // ry_q0(a2): rows i: (xi0, xi1) -> (c*xi0 - s*xi1, s*xi0 + c*xi1)
    t0 = c2 * x00 - s2 * x01; t1 = s2 * x00 + c2 * x01; x00 = t0; x01 = t1;
    t0 = c2 * x10 - s2 * x11; t1 = s2 * x10 + c2 * x11; x10 = t0; x11 = t1;

    // ry_q1(a3)
    t0 = c3 * x00 - s3 * x10; t1 = s3 * x00 + c3 * x10; x00 = t0; x10 = t1;
    t0 = c3 * x01 - s3 * x11; t1 = s3 * x01 + c3 * x11; x01 = t0; x11 = t1;

    // ry_q0(theta0)
    t0 = ct0 * x00 - st0 * x01; t1 = st0 * x00 + ct0 * x01; x00 = t0; x01 = t1;
    t0 = ct0 * x10 - st0 * x11; t1 = st0 * x10 + ct0 * x11; x10 = t0; x11 = t1;

    // ry_q1(theta1)
    t0 = ct1 * x00 - st1 * x10; t1 = st1 * x00 + ct1 * x10; x00 = t0; x10 = t1;
    t0 = ct1 * x01 - st1 * x11; t1 = st1 * x01 + ct1 * x11; x01 = t0; x11 = t1;

    return (x00 * x00 + x10 * x10) - (x01 * x01 + x11 * x11);
}

// ---------------------------------------------------------------------------
// Main kernel: double-buffered async global->LDS streaming pipeline (CDNA5
// GLOBAL_LOAD_ASYNC_TO_LDS_B128 + S_WAIT_ASYNCCNT). Each thread owns one
// 16-byte LDS slot per buffer; ASYNCcnt is per-wave and async loads complete
// in order, so no workgroup barriers are needed.
// ---------------------------------------------------------------------------
__launch_bounds__(TPB)
__global__ void qconv_stream_kernel(const float4* __restrict__ patches,
                                    const float* __restrict__ theta,
                                    float* __restrict__ out, int nTiles) {
    __shared__ float4 buf[2][TPB];   // 8 KB of the 320 KB/WGP LDS

    // theta rotation is shared by all patches: evaluate once per thread.
    const float th0 = theta[0];
    const float th1 = theta[1];
    const float ct0 = __cosf(0.5f * th0), st0 = __sinf(0.5f * th0);
    const float ct1 = __cosf(0.5f * th1), st1 = __sinf(0.5f * th1);

    const unsigned tid = threadIdx.x;

    auto issue_async = [&](int tile, int b) {
        // Low 32 bits of a generic pointer into LDS == offset within the
        // wave's LDS allocation (flat LDS aperture layout), which is exactly
        // the VDST operand of global_load_async_to_lds.
        uint32_t lds_off = (uint32_t)(uintptr_t)(&buf[b][tid]);
        uint64_t gaddr   = (uint64_t)(uintptr_t)(patches + (size_t)tile * TPB + tid);
        asm volatile("global_load_async_to_lds_b128 %0, %1, off"
                     :: "v"(lds_off), "v"(gaddr)
                     : "memory");
    };

    int tile0 = blockIdx.x;
    if (tile0 >= nTiles) return;

    issue_async(tile0, 0);                          // prime the pipe (ASYNCcnt=1)

    int k = 0;
    for (int tile = tile0; tile < nTiles; tile += gridDim.x, ++k) {
        int nextTile = tile + gridDim.x;
        if (nextTile < nTiles) {
            // Make sure our ds_loads of the buffer we are about to overwrite
            // have drained, then prefetch the next tile into it.
            asm volatile("s_wait_dscnt 0" ::: "memory");
            issue_async(nextTile, (k + 1) & 1);
            // Two async ops outstanding; loads retire in order, so <=1 means
            // the current tile's data is in LDS.
            asm volatile("s_wait_asynccnt 1" ::: "memory");
        } else {
            asm volatile("s_wait_asynccnt 0" ::: "memory");
        }

        float4 p = buf[k & 1][tid];                 // ds_load_b128
        float r = qsim(p.x, p.y, p.z, p.w, ct0, st0, ct1, st1);
        out[(size_t)tile * TPB + tid] = r;          // coalesced b32 store
    }
}

// Tail for N % 256 != 0 (not hit for N = 2^23, kept for generality).
__global__ void qconv_tail_kernel(const float* __restrict__ patches,
                                  const float* __restrict__ theta,
                                  float* __restrict__ out, int start, int n) {
    int i = start + blockIdx.x * blockDim.x + threadIdx.x;
    if (i >= n) return;
    const float ct0 = __cosf(0.5f * theta[0]), st0 = __sinf(0.5f * theta[0]);
    const float ct1 = __cosf(0.5f * theta[1]), st1 = __sinf(0.5f * theta[1]);
    float a0 = patches[4 * (size_t)i + 0];
    float a1 = patches[4 * (size_t)i + 1];
    float a2 = patches[4 * (size_t)i + 2];
    float a3 = patches[4 * (size_t)i + 3];
    out[i] = qsim(a0, a1, a2, a3, ct0, st0, ct1, st1);
}

extern "C" void kernel_launch(void* const* d_in, const int* in_sizes, int n_in,
                              void* d_out, int out_size, void* d_ws, size_t ws_size,
                              hipStream_t stream) {
    (void)n_in; (void)d_ws; (void)ws_size; (void)out_size;
    const float* patches = (const float*)d_in[0];   // (N,4) f32, row-major
    const float* theta   = (const float*)d_in[1];   // (2,)  f32
    float*       out     = (float*)d_out;           // (N,)  f32

    int n      = in_sizes[0] / 4;
    int nTiles = n / TPB;

    if (nTiles > 0) {
        int blocks = nTiles < 4096 ? nTiles : 4096;  // ~8 tiles/block at N=2^23
        qconv_stream_kernel<<<blocks, TPB, 0, stream>>>(
            (const float4*)patches, theta, out, nTiles);
    }
    int rem = n - nTiles * TPB;
    if (rem > 0) {
        qconv_tail_kernel<<<(rem + TPB - 1) / TPB, TPB, 0, stream>>>(
            patches, theta, out, nTiles * TPB, n);
    }
}